// SimpleRetention_26130581029009
// MI455X (gfx1250) — compile-verified
//
#include <hip/hip_runtime.h>

// ---------------------------------------------------------------------------
// SimpleRetention on MI455X (gfx1250), fp32 end-to-end via V_WMMA_F32_16X16X4_F32.
//
//   Q = xpos(X @ W_Q), K = xpos_inv(X @ W_K), V = X @ W_V
//   out = ((Q K^T) * gamma^(n-m) [causal]) @ V
//
// Kernel 1: projection GEMMs (K=256 -> 64 wmma per 16x16 tile, 2 independent
//           accumulator chains) + xPos in C-layout registers (partner column
//           via __shfl_xor on wave32).
// Kernel 2: flash-style causal retention. 4 waves per block split the key
//           range of one 16-row query tile (4096 wave32s total), each stages
//           its decayed S tile through wave-private padded LDS (C->A layout
//           conversion), accumulates O, then an LDS tree-reduction merges the
//           4 partials. gamma^(n-m) costs one __expf per key tile (separable
//           decay), mask select only on the diagonal tile.
// Exact-window optimization: gamma^D underflows fp32 for D >= ~3264, so the
// reference's own (fp32) D matrix is exactly 0 there; skipping tiles older
// than 213 is bit-equivalent.
// ---------------------------------------------------------------------------

typedef __attribute__((ext_vector_type(2))) float v2f;
typedef __attribute__((ext_vector_type(8))) float v8f;

#define BATCH  4
#define SEQ    4096
#define HIDDEN 256
#define HEAD   64
#define VDIM   64
#define NT_SEQ (SEQ / 16)
#define WAVES  4
#define WIN_TILES 213

constexpr float LN_GAMMA = -0.03174869831f;  // ln(31/32)
constexpr float LN_10000 =  9.21034037198f;

// D = A(16x4) * B(4x16) + C, fp32.
// A frag (per lane): M = lane%16, K = {2*(lane/16), 2*(lane/16)+1}
// B frag (per lane): N = lane%16, K = {2*(lane/16), 2*(lane/16)+1} across 2 VGPRs
// C/D:  VGPR r, lanes 0-15 -> (M=r, N=lane), lanes 16-31 -> (M=r+8, N=lane-16)
__device__ inline v8f wmma_f32(v2f a, v2f b, v8f c) {
    return __builtin_amdgcn_wmma_f32_16x16x4_f32(false, a, false, b, (short)0, c,
                                                 false, false);
}

// ---------------------------------------------------------------------------
// Kernel 1: Q/K/V projection + xPos.
// grid.x = BATCH*NT_SEQ (16-row tiles), grid.y = 12 (3 matrices x 4 col tiles)
// ---------------------------------------------------------------------------
__global__ __launch_bounds__(32)
void proj_xpos_kernel(const float* __restrict__ X,
                      const float* __restrict__ WQ,
                      const float* __restrict__ WK,
                      const float* __restrict__ WV,
                      float* __restrict__ Qo,
                      float* __restrict__ Ko,
                      float* __restrict__ Vo)
{
    const int lane = threadIdx.x;
    const int col  = lane & 15;
    const int hi   = lane >> 4;

    const int rt   = blockIdx.x;
    const int b    = rt / NT_SEQ;
    const int row0 = (rt % NT_SEQ) * 16;

    const int which = blockIdx.y >> 2;   // 0=Q, 1=K, 2=V
    const int nt    = blockIdx.y & 3;
    const int n0    = nt * 16;

    const float* __restrict__ W = (which == 0) ? WQ : (which == 1) ? WK : WV;
    float*       __restrict__ Y = (which == 0) ? Qo : (which == 1) ? Ko : Vo;

    const float* __restrict__ Xrow =
        X + ((size_t)b * SEQ + row0 + col) * HIDDEN + 2 * hi;

    // Two independent accumulator chains to break WMMA RAW dependency.
    v8f acc0 = {}, acc1 = {};
    #pragma unroll 4
    for (int kb = 0; kb < HIDDEN; kb += 8) {
        v2f a0 = *(const v2f*)(Xrow + kb);
        v2f a1 = *(const v2f*)(Xrow + kb + 4);
        v2f b0, b1;
        b0.x = W[(size_t)(kb + 2 * hi + 0) * HEAD + n0 + col];
        b0.y = W[(size_t)(kb + 2 * hi + 1) * HEAD + n0 + col];
        b1.x = W[(size_t)(kb + 2 * hi + 4) * HEAD + n0 + col];
        b1.y = W[(size_t)(kb + 2 * hi + 5) * HEAD + n0 + col];
        acc0 = wmma_f32(a0, b0, acc0);
        acc1 = wmma_f32(a1, b1, acc1);
    }
    v8f acc = acc0 + acc1;

    if (which < 2) {
        // xPos: output column fixed per lane; partner column (dcol^1) lives in
        // the adjacent lane -> __shfl_xor on wave32.
        const int   dcol    = n0 + col;
        const int   p       = dcol >> 1;
        const float invf    = __expf(-((float)p * (1.0f / 32.0f)) * LN_10000);
        const float lnscale = __logf(((float)(2 * p) + 25.6f) * (1.0f / 89.6f));
        const float kexp    = ((which == 1) ? -lnscale : lnscale) * (1.0f / 512.0f);
        #pragma unroll
        for (int r = 0; r < 8; ++r) {
            const int   n   = row0 + r + 8 * hi;
            const float ang = (float)n * invf;
            const float sv  = __sinf(ang);
            const float cv  = __cosf(ang);
            const float sc  = __expf((float)n * kexp);
            const float x   = acc[r];
            const float px  = __shfl_xor(x, 1, 32);
            const float rot = (dcol & 1) ? px : -px;           // (-x2, x1) pattern
            acc[r] = x * (cv * sc) + rot * (sv * sc);
        }
    }

    #pragma unroll
    for (int r = 0; r < 8; ++r) {
        Y[((size_t)b * SEQ + row0 + r + 8 * hi) * HEAD + n0 + col] = acc[r];
    }
}

// ---------------------------------------------------------------------------
// Kernel 2: causal retention. Block = 4 waves sharing one 16-row query tile,
// each wave owns a contiguous chunk of the key-tile window.
// ---------------------------------------------------------------------------
__global__ __launch_bounds__(WAVES * 32)
void retention_kernel(const float* __restrict__ Q,
                      const float* __restrict__ K,
                      const float* __restrict__ V,
                      float* __restrict__ O)
{
    __shared__ float stile[WAVES][16][17];    // wave-private S staging (padded)
    __shared__ float red[WAVES][16][VDIM];    // per-wave O partials

    const int tid  = threadIdx.x;
    const int wave = tid >> 5;
    const int lane = tid & 31;
    const int col  = lane & 15;
    const int hi   = lane >> 4;

    const int rt   = blockIdx.x;
    const int b    = rt / NT_SEQ;
    const int it   = rt % NT_SEQ;
    const int row0 = it * 16;

    // Key-tile window [j0, it], split into 4 contiguous chunks.
    const int j0  = (it >= WIN_TILES) ? (it - WIN_TILES + 1) : 0;
    const int len = it - j0 + 1;
    const int lo  = j0 + (len * wave) / WAVES;
    const int hiE = j0 + (len * (wave + 1)) / WAVES;   // exclusive

    // Q A-fragments resident for the whole chunk (16 x float2 = 32 VGPRs).
    const float* __restrict__ Qrow =
        Q + ((size_t)b * SEQ + row0 + col) * HEAD + 2 * hi;
    v2f qf[16];
    #pragma unroll
    for (int kk = 0; kk < 16; ++kk) qf[kk] = *(const v2f*)(Qrow + 4 * kk);

    // Separable decay: gamma^(n-m) = gamma^(row0-m0) * gamma^((r+8hi)-col).
    // Second factor is loop-invariant per lane.
    float wl[8];
    #pragma unroll
    for (int r = 0; r < 8; ++r)
        wl[r] = __expf((float)(r + 8 * hi - col) * LN_GAMMA);

    v8f acc[4] = {v8f{}, v8f{}, v8f{}, v8f{}};

    for (int jt = lo; jt < hiE; ++jt) {
        const int m0 = jt * 16;

        if (jt + 1 < hiE) {   // pull next key tile (global_prefetch_b8)
            __builtin_prefetch(K + ((size_t)b * SEQ + m0 + 16 + col) * HEAD, 0, 1);
            __builtin_prefetch(V + ((size_t)b * SEQ + m0 + 16 + col) * VDIM, 0, 1);
        }

        // ---- S = Q . K^T, two independent accumulator chains.
        const float* __restrict__ Krow =
            K + ((size_t)b * SEQ + m0 + col) * HEAD + 2 * hi;
        v8f s0 = {}, s1 = {};
        #pragma unroll
        for (int kk = 0; kk < 16; kk += 2) {
            v2f b0 = *(const v2f*)(Krow + 4 * kk);
            v2f b1 = *(const v2f*)(Krow + 4 * kk + 4);
            s0 = wmma_f32(qf[kk],     b0, s0);
            s1 = wmma_f32(qf[kk + 1], b1, s1);
        }
        v8f s = s0 + s1;

        // ---- decay: one __expf per tile; mask only matters on the diagonal.
        const float tf = __expf((float)(row0 - m0) * LN_GAMMA);
        const bool diag = (jt == it);
        #pragma unroll
        for (int r = 0; r < 8; ++r) {
            float w = tf * wl[r];
            if (diag && (r + 8 * hi) < col) w = 0.0f;
            stile[wave][r + 8 * hi][col] = s[r] * w;
        }
        // Wave-private LDS; same-wave DS ops are in-order (ISA 7.3) -> a
        // compiler/memory fence suffices. NO barrier here: waves have
        // different trip counts, __syncthreads would be illegal.
        __builtin_amdgcn_fence(__ATOMIC_ACQ_REL, "workgroup");

        // ---- O += S . V  (K-dim = m in 4 steps of 4; 4 output col tiles)
        #pragma unroll
        for (int kk = 0; kk < 4; ++kk) {
            v2f a;
            a.x = stile[wave][col][4 * kk + 2 * hi + 0];
            a.y = stile[wave][col][4 * kk + 2 * hi + 1];
            const float* __restrict__ Vr =
                V + ((size_t)b * SEQ + m0 + 4 * kk + 2 * hi) * VDIM + col;
            #pragma unroll
            for (int vt = 0; vt < 4; ++vt) {
                v2f bm;
                bm.x = Vr[vt * 16];
                bm.y = Vr[VDIM + vt * 16];
                acc[vt] = wmma_f32(a, bm, acc[vt]);
            }
        }
        __builtin_amdgcn_fence(__ATOMIC_ACQ_REL, "workgroup");  // WAR on stile
    }

    // ---- publish per-wave partials, then block-wide reduction.
    #pragma unroll
    for (int vt = 0; vt < 4; ++vt)
        #pragma unroll
        for (int r = 0; r < 8; ++r)
            red[wave][r + 8 * hi][vt * 16 + col] = acc[vt][r];

    __syncthreads();   // uniform: every wave reaches this exactly once

    // 128 threads reduce 16x64 outputs: 8 consecutive columns per thread.
    const int orow = tid >> 3;
    const int oc0  = (tid & 7) * 8;
    float* __restrict__ Orow =
        O + ((size_t)b * SEQ + row0 + orow) * VDIM + oc0;
    #pragma unroll
    for (int k = 0; k < 8; ++k) {
        float v = red[0][orow][oc0 + k] + red[1][orow][oc0 + k] +
                  red[2][orow][oc0 + k] + red[3][orow][oc0 + k];
        Orow[k] = v;
    }
}

// ---------------------------------------------------------------------------
extern "C" void kernel_launch(void* const* d_in, const int* in_sizes, int n_in,
                              void* d_out, int out_size, void* d_ws, size_t ws_size,
                              hipStream_t stream)
{
    const float* X  = (const float*)d_in[0];
    const float* WQ = (const float*)d_in[1];
    const float* WK = (const float*)d_in[2];
    const float* WV = (const float*)d_in[3];

    // Workspace: Q, K, V projections (3 x B*L*64 fp32 = 12 MB).
    float* Qw = (float*)d_ws;
    float* Kw = Qw + (size_t)BATCH * SEQ * HEAD;
    float* Vw = Kw + (size_t)BATCH * SEQ * HEAD;

    dim3 gProj(BATCH * NT_SEQ, 12);   // 12288 wave32s
    proj_xpos_kernel<<<gProj, 32, 0, stream>>>(X, WQ, WK, WV, Qw, Kw, Vw);

    retention_kernel<<<BATCH * NT_SEQ, WAVES * 32, 0, stream>>>(Qw, Kw, Vw,
                                                                (float*)d_out);
}